// DecoderNet_6691559047526
// MI455X (gfx1250) — compile-verified
//
#include <hip/hip_runtime.h>
#include <math.h>

// ---------------------------------------------------------------------------
// DecoderNet greedy decode, algebraically simplified:
//   * attention MLP is fully linear -> h-term cancels in softmax -> ctx is
//     step-invariant; computed once from u = W1_top @ W2 @ W3 @ W4 @ Ww
//   * gctx = ctx @ W_ih[:,WD:]^T + b_ih precomputed once
//   * per step: one fused kernel for gi = emb[w]@W_ih[:,:WD]^T + gctx and
//               gh = h@W_hh^T + b_hh (blockIdx.z selects), GRU elementwise,
//               logits = h'@Wo + bo (written straight into d_out), argmax.
// All GEMMs use V_WMMA_F32_16X16X4_F32 (exact fp32) on wave32, with a
// 4-stage register pipeline for B fragments (16 loads in flight per wave)
// and two-phase b128 staging of the A tile into LDS.
// ---------------------------------------------------------------------------

typedef float v2f __attribute__((ext_vector_type(2)));
typedef float v8f __attribute__((ext_vector_type(8)));

#define HSZ   512
#define WDIM  1024
#define VOC   4096
#define BATCH 256
#define SLEN  80
#define NSTEP 27
#define G3    1536   // 3*H

// ---- one-time: u = W1[0:H,:] @ (W2 @ (W3 @ (W4 @ Ww))) ---------------------
__global__ void chain_kernel(const float* __restrict__ W1, const float* __restrict__ W2,
                             const float* __restrict__ W3, const float* __restrict__ W4,
                             const float* __restrict__ Ww, float* __restrict__ u) {
  __shared__ float t[HSZ];
  __shared__ float tn[HSZ];
  int tid = threadIdx.x;                       // 512 threads
  t[tid] = Ww[tid];
  __syncthreads();
  const float* Ws[3] = {W4, W3, W2};
  for (int l = 0; l < 3; ++l) {
    const float* Wm = Ws[l];
    float acc = 0.f;
    for (int k = 0; k < HSZ; ++k) acc += Wm[tid * HSZ + k] * t[k];
    tn[tid] = acc;  __syncthreads();
    t[tid] = tn[tid]; __syncthreads();
  }
  float acc = 0.f;
  for (int k = 0; k < HSZ; ++k) acc += W1[tid * HSZ + k] * t[k];
  u[tid] = acc;
}

// ---- one-time: scores=enc.u, softmax over S, ctx = sum_s attw*enc ----------
__global__ void attn_ctx_kernel(const float* __restrict__ enc, const float* __restrict__ u,
                                float* __restrict__ ctx) {
  __shared__ float sc[SLEN];
  __shared__ float mx, sm;
  int b = blockIdx.x, tid = threadIdx.x;
  int lane = tid & 31, wave = tid >> 5;        // 8 waves of 32 (wave32)
  const float* eb = enc + (size_t)b * SLEN * HSZ;
  for (int s = wave; s < SLEN; s += 8) {
    float p = 0.f;
    for (int k = lane; k < HSZ; k += 32) p += eb[s * HSZ + k] * u[k];
    for (int off = 16; off > 0; off >>= 1) p += __shfl_down(p, off, 32);
    if (lane == 0) sc[s] = p;
  }
  __syncthreads();
  if (tid == 0) { float m = sc[0]; for (int s = 1; s < SLEN; ++s) m = fmaxf(m, sc[s]); mx = m; }
  __syncthreads();
  if (tid < SLEN) sc[tid] = expf(sc[tid] - mx);
  __syncthreads();
  if (tid == 0) { float t = 0.f; for (int s = 0; s < SLEN; ++s) t += sc[s]; sm = t; }
  __syncthreads();
  float inv = 1.f / sm;
  for (int h = tid; h < HSZ; h += 256) {
    float a = 0.f;
    for (int s = 0; s < SLEN; ++s) a += sc[s] * eb[s * HSZ + h];
    ctx[(size_t)b * HSZ + h] = a * inv;
  }
}

// ---- GEMM flavor T core: C[m,n] = sum_k A[m,k]*W[n*ldw+wOff+k] -------------
// One wave per 16x16 tile * 4 subtiles; 8 waves/block share a 16x512 A-chunk
// in LDS (two-phase b128 staging). B fragments use a 4-stage register
// pipeline (prefetch distance 16 k-steps => 16 loads in flight per wave).
__device__ __forceinline__ void
gemm_awt_dev(const float* __restrict__ A, int lda, const int* __restrict__ gidx,
             const float* __restrict__ W, int ldw, int wOff,
             const float* __restrict__ addM, const float* __restrict__ biasJ,
             float* __restrict__ C, int ldc, int K, int N,
             int mBase, int colBase) {
  __shared__ float At[16 * 512];
  const int tid = threadIdx.x, lane = tid & 31;
  const int half = lane >> 4;                  // K-pair select per ISA A layout
  const int l16 = lane & 15;                   // M row / N col within tile

  const float* wp[4];
#pragma unroll
  for (int t = 0; t < 4; ++t)
    wp[t] = W + (size_t)(colBase + t * 16 + l16) * ldw + wOff + 2 * half;

  v8f acc[4] = {};
  for (int k0 = 0; k0 < K; k0 += 512) {
    if (k0) __syncthreads();
    // two-phase staging: 8 b128 loads in flight, then 8 ds_store_b128
    float4* At4 = reinterpret_cast<float4*>(At);
    float4 tmp[8];
#pragma unroll
    for (int j = 0; j < 8; ++j) {
      int i = tid + j * 256;
      int m = i >> 7, k4 = (i & 127) << 2;
      int row = gidx ? gidx[mBase + m] : (mBase + m);
      tmp[j] = *reinterpret_cast<const float4*>(A + (unsigned)(row * lda + k0 + k4));
    }
#pragma unroll
    for (int j = 0; j < 8; ++j) At4[tid + j * 256] = tmp[j];
    __syncthreads();

    // 4-stage B pipeline
    v2f bq[4][4];
#pragma unroll
    for (int s = 0; s < 4; ++s)
#pragma unroll
      for (int t = 0; t < 4; ++t) {
        bq[s][t].x = wp[t][k0 + 4 * s];
        bq[s][t].y = wp[t][k0 + 4 * s + 1];
      }
    int kk = 0;
    for (; kk < 512 - 16; kk += 16) {
#pragma unroll
      for (int s = 0; s < 4; ++s) {
        int kc = kk + 4 * s;
        v2f bcur[4];
#pragma unroll
        for (int t = 0; t < 4; ++t) bcur[t] = bq[s][t];
#pragma unroll
        for (int t = 0; t < 4; ++t) {          // reload 16 k-steps ahead
          bq[s][t].x = wp[t][k0 + kc + 16];
          bq[s][t].y = wp[t][k0 + kc + 17];
        }
        v2f a;
        int ai = l16 * 512 + kc + 2 * half;
        a.x = At[ai]; a.y = At[ai + 1];
#pragma unroll
        for (int t = 0; t < 4; ++t)
          acc[t] = __builtin_amdgcn_wmma_f32_16x16x4_f32(
              false, a, false, bcur[t], (short)0, acc[t], false, false);
      }
    }
    // peeled last 16-k group: drain pipeline, no reloads
#pragma unroll
    for (int s = 0; s < 4; ++s) {
      int kc = kk + 4 * s;
      v2f a;
      int ai = l16 * 512 + kc + 2 * half;
      a.x = At[ai]; a.y = At[ai + 1];
#pragma unroll
      for (int t = 0; t < 4; ++t)
        acc[t] = __builtin_amdgcn_wmma_f32_16x16x4_f32(
            false, a, false, bq[s][t], (short)0, acc[t], false, false);
    }
  }
#pragma unroll
  for (int t = 0; t < 4; ++t) {
    int gn = colBase + t * 16 + l16;
    if (gn < N) {
#pragma unroll
      for (int i = 0; i < 8; ++i) {
        int gm = mBase + i + 8 * half;         // C/D layout: VGPR i -> row i+8*half
        float v = acc[t][i];
        if (addM)  v += addM[(size_t)gm * ldc + gn];
        if (biasJ) v += biasJ[gn];
        C[(size_t)gm * ldc + gn] = v;
      }
    }
  }
}

// standalone flavor-T kernel (used for gctx)
__global__ void __launch_bounds__(256)
gemm_awt_kernel(const float* __restrict__ A, int lda, const int* __restrict__ gidx,
                const float* __restrict__ W, int ldw, int wOff,
                const float* __restrict__ addM, const float* __restrict__ biasJ,
                float* __restrict__ C, int ldc, int K, int N) {
  gemm_awt_dev(A, lda, gidx, W, ldw, wOff, addM, biasJ, C, ldc, K, N,
               blockIdx.x * 16, blockIdx.y * 512 + (threadIdx.x >> 5) * 64);
}

// fused per-step gates kernel: z==0 -> gi (gathered emb), z==1 -> gh
__global__ void __launch_bounds__(256)
gates_kernel(const float* __restrict__ emb, const int* __restrict__ wCur,
             const float* __restrict__ W_ih, const float* __restrict__ gctx,
             const float* __restrict__ hPrev, const float* __restrict__ W_hh,
             const float* __restrict__ b_hh,
             float* __restrict__ gi, float* __restrict__ gh) {
  int mBase = blockIdx.x * 16;
  int colBase = blockIdx.y * 512 + (threadIdx.x >> 5) * 64;
  if (blockIdx.z == 0) {
    gemm_awt_dev(emb, WDIM, wCur, W_ih, G3, 0, gctx, nullptr,
                 gi, G3, WDIM, G3, mBase, colBase);
  } else {
    gemm_awt_dev(hPrev, HSZ, nullptr, W_hh, HSZ, 0, nullptr, b_hh,
                 gh, G3, HSZ, G3, mBase, colBase);
  }
}

// ---- GEMM flavor B: C[m,n] = sum_k A[m,k]*Bm[k*ldb+n] + biasJ (logits) -----
__global__ void __launch_bounds__(256)
gemm_ab_kernel(const float* __restrict__ A, int lda,
               const float* __restrict__ Bm, int ldb,
               const float* __restrict__ biasJ,
               float* __restrict__ C, long long ldc, int K, int N) {
  __shared__ float At[16 * 512];
  const int tid = threadIdx.x, lane = tid & 31;
  const int wave = tid >> 5;
  const int mBase = blockIdx.x * 16;
  const int colBase = blockIdx.y * 512 + wave * 64;
  const int half = lane >> 4;
  const int l16 = lane & 15;

  const float* bp[4];
#pragma unroll
  for (int t = 0; t < 4; ++t) bp[t] = Bm + colBase + t * 16 + l16;

  v8f acc[4] = {};
  for (int k0 = 0; k0 < K; k0 += 512) {
    if (k0) __syncthreads();
    float4* At4 = reinterpret_cast<float4*>(At);
    float4 tmp[8];
#pragma unroll
    for (int j = 0; j < 8; ++j) {
      int i = tid + j * 256;
      int m = i >> 7, k4 = (i & 127) << 2;
      tmp[j] = *reinterpret_cast<const float4*>(A + (unsigned)((mBase + m) * lda + k0 + k4));
    }
#pragma unroll
    for (int j = 0; j < 8; ++j) At4[tid + j * 256] = tmp[j];
    __syncthreads();

    v2f bq[4][4];
#pragma unroll
    for (int s = 0; s < 4; ++s)
#pragma unroll
      for (int t = 0; t < 4; ++t) {
        bq[s][t].x = bp[t][(unsigned)((k0 + 4 * s + 2 * half) * ldb)];
        bq[s][t].y = bp[t][(unsigned)((k0 + 4 * s + 2 * half + 1) * ldb)];
      }
    int kk = 0;
    for (; kk < 512 - 16; kk += 16) {
      __builtin_prefetch(bp[0] + (unsigned)((k0 + kk + 64) * ldb), 0, 1);
#pragma unroll
      for (int s = 0; s < 4; ++s) {
        int kc = kk + 4 * s;
        v2f bcur[4];
#pragma unroll
        for (int t = 0; t < 4; ++t) bcur[t] = bq[s][t];
        int kr = k0 + kc + 16 + 2 * half;
#pragma unroll
        for (int t = 0; t < 4; ++t) {          // reload 16 k-steps ahead
          bq[s][t].x = bp[t][(unsigned)(kr * ldb)];
          bq[s][t].y = bp[t][(unsigned)((kr + 1) * ldb)];
        }
        v2f a;
        int ai = l16 * 512 + kc + 2 * half;
        a.x = At[ai]; a.y = At[ai + 1];
#pragma unroll
        for (int t = 0; t < 4; ++t)
          acc[t] = __builtin_amdgcn_wmma_f32_16x16x4_f32(
              false, a, false, bcur[t], (short)0, acc[t], false, false);
      }
    }
#pragma unroll
    for (int s = 0; s < 4; ++s) {              // drain, no reloads
      int kc = kk + 4 * s;
      v2f a;
      int ai = l16 * 512 + kc + 2 * half;
      a.x = At[ai]; a.y = At[ai + 1];
#pragma unroll
      for (int t = 0; t < 4; ++t)
        acc[t] = __builtin_amdgcn_wmma_f32_16x16x4_f32(
            false, a, false, bq[s][t], (short)0, acc[t], false, false);
    }
  }
#pragma unroll
  for (int t = 0; t < 4; ++t) {
    int gn = colBase + t * 16 + l16;
    if (gn < N) {
#pragma unroll
      for (int i = 0; i < 8; ++i) {
        int gm = mBase + i + 8 * half;
        float v = acc[t][i] + (biasJ ? biasJ[gn] : 0.f);
        C[(size_t)gm * ldc + gn] = v;
      }
    }
  }
}

// ---- GRU elementwise --------------------------------------------------------
__global__ void gru_kernel(const float* __restrict__ gi, const float* __restrict__ gh,
                           const float* __restrict__ hPrev, float* __restrict__ hNext) {
  int i = blockIdx.x * 256 + threadIdx.x;      // < BATCH*H
  int b = i >> 9, j = i & 511;
  const float* gib = gi + (size_t)b * G3;
  const float* ghb = gh + (size_t)b * G3;
  float ir = gib[j], iz = gib[HSZ + j], in_ = gib[2 * HSZ + j];
  float hr = ghb[j], hz = ghb[HSZ + j], hn = ghb[2 * HSZ + j];
  float r = 1.f / (1.f + expf(-(ir + hr)));
  float z = 1.f / (1.f + expf(-(iz + hz)));
  float n = tanhf(in_ + r * hn);
  hNext[i] = (1.f - z) * n + z * hPrev[i];
}

// ---- per-row argmax over V (first-max tie break, matches jnp.argmax) -------
__global__ void argmax_kernel(const float* __restrict__ logits, long long rowStride,
                              float* __restrict__ predOut, int predStride,
                              int* __restrict__ wNext) {
  __shared__ float bv[256];
  __shared__ int bi[256];
  int b = blockIdx.x, tid = threadIdx.x;
  const float* row = logits + (size_t)b * rowStride;
  float best = -INFINITY; int bidx = 0x7fffffff;
  for (int n = tid; n < VOC; n += 256) {
    float v = row[n];
    if (v > best) { best = v; bidx = n; }
  }
  bv[tid] = best; bi[tid] = bidx;
  __syncthreads();
  for (int off = 128; off > 0; off >>= 1) {
    if (tid < off) {
      if (bv[tid + off] > bv[tid] ||
          (bv[tid + off] == bv[tid] && bi[tid + off] < bi[tid])) {
        bv[tid] = bv[tid + off]; bi[tid] = bi[tid + off];
      }
    }
    __syncthreads();
  }
  if (tid == 0) { wNext[b] = bi[0]; predOut[(size_t)b * predStride] = (float)bi[0]; }
}

// ---- init: h0 = encoder_last_hidden_state, w0 = <BOS>=1 --------------------
__global__ void init_kernel(const float* __restrict__ elhs, float* __restrict__ hCur,
                            int* __restrict__ wCur) {
  int i = blockIdx.x * 256 + threadIdx.x;
  if (i < BATCH * HSZ) hCur[i] = elhs[i];
  if (i < BATCH) wCur[i] = 1;
}

// ---------------------------------------------------------------------------
extern "C" void kernel_launch(void* const* d_in, const int* in_sizes, int n_in,
                              void* d_out, int out_size, void* d_ws, size_t ws_size,
                              hipStream_t stream) {
  (void)in_sizes; (void)n_in; (void)out_size; (void)ws_size;
  const float* elhs = (const float*)d_in[0];
  const float* enc  = (const float*)d_in[1];
  /* d_in[2] targets: only its length (T) matters; NSTEP = T-1 = 27 */
  const float* emb  = (const float*)d_in[3];
  const float* W1   = (const float*)d_in[4];
  const float* W2   = (const float*)d_in[6];
  const float* W3   = (const float*)d_in[8];
  const float* W4   = (const float*)d_in[10];
  const float* Ww   = (const float*)d_in[12];
  const float* W_ih = (const float*)d_in[13];
  const float* b_ih = (const float*)d_in[14];
  const float* W_hh = (const float*)d_in[15];
  const float* b_hh = (const float*)d_in[16];
  const float* Wo   = (const float*)d_in[17];
  const float* bo   = (const float*)d_in[18];

  float* ws   = (float*)d_ws;
  float* u    = ws;                       // 512
  float* ctx  = u + HSZ;                  // B*H
  float* gctx = ctx + BATCH * HSZ;        // B*3H
  float* gi   = gctx + BATCH * G3;        // B*3H
  float* gh   = gi + BATCH * G3;          // B*3H
  float* hA   = gh + BATCH * G3;          // B*H
  float* hB   = hA + BATCH * HSZ;         // B*H
  int*   w0   = (int*)(hB + BATCH * HSZ); // B
  int*   w1   = w0 + BATCH;               // B

  float* out      = (float*)d_out;                       // (B, 27, V) logits
  float* predBase = out + (size_t)BATCH * NSTEP * VOC;   // (B, 27) predictions

  // One-time (per call) precompute.
  chain_kernel<<<1, 512, 0, stream>>>(W1, W2, W3, W4, Ww, u);
  attn_ctx_kernel<<<BATCH, 256, 0, stream>>>(enc, u, ctx);
  // gctx = ctx @ W_ih[:, WD:]^T + b_ih   (M=256, N=1536, K=512)
  gemm_awt_kernel<<<dim3(BATCH / 16, G3 / 512), 256, 0, stream>>>(
      ctx, HSZ, nullptr, W_ih, G3, WDIM, nullptr, b_ih, gctx, G3, HSZ, G3);
  init_kernel<<<(BATCH * HSZ + 255) / 256, 256, 0, stream>>>(elhs, hA, w0);

  for (int t = 0; t < NSTEP; ++t) {
    float* hPrev = (t & 1) ? hB : hA;
    float* hNext = (t & 1) ? hA : hB;
    int* wCur = (t & 1) ? w1 : w0;
    int* wNxt = (t & 1) ? w0 : w1;

    // fused: z=0 -> gi = embedding[wCur]@W_ih[:,:WD]^T + gctx (K=1024, gather)
    //        z=1 -> gh = hPrev@W_hh^T + b_hh                  (K=512)
    gates_kernel<<<dim3(BATCH / 16, G3 / 512, 2), 256, 0, stream>>>(
        emb, wCur, W_ih, gctx, hPrev, W_hh, b_hh, gi, gh);
    gru_kernel<<<BATCH * HSZ / 256, 256, 0, stream>>>(gi, gh, hPrev, hNext);
    // logits = hNext @ Wo + bo -> d_out[(b*27 + t)*V + n]
    gemm_ab_kernel<<<dim3(BATCH / 16, VOC / 512), 256, 0, stream>>>(
        hNext, HSZ, Wo, VOC, bo, out + (size_t)t * VOC, (long long)NSTEP * VOC, HSZ, VOC);
    argmax_kernel<<<BATCH, 256, 0, stream>>>(
        out + (size_t)t * VOC, (long long)NSTEP * VOC, predBase + t, NSTEP, wNxt);
  }
}